// RGCNModel_39505109188791
// MI455X (gfx1250) — compile-verified
//
#include <hip/hip_runtime.h>
#include <hip/hip_bf16.h>

// Problem constants (match reference)
#define NN 100000
#define FF 128
#define RR 4
#define EE 1600000
#define LL 2

typedef __attribute__((ext_vector_type(2))) float v2f;
typedef __attribute__((ext_vector_type(8))) float v8f;

// ---------------------------------------------------------------------------
// Phase 1: per-edge scatter of raw features into (dst, relation) buckets.
// mean-aggregation commutes with the linear relation transform, so we
// aggregate x[src] first and do the GEMM afterwards (saves a [R,N,F] tensor).
// One wave (32 lanes) per edge; each lane handles 4 consecutive features.
// ---------------------------------------------------------------------------
__global__ __launch_bounds__(256)
void rgcn_scatter(const float* __restrict__ x,
                  const int* __restrict__ src,
                  const int* __restrict__ dst,
                  const int* __restrict__ etype,
                  float* __restrict__ agg,   // [N, R, F]
                  float* __restrict__ cnt)   // [N, R]
{
    const int wave = (blockIdx.x * blockDim.x + threadIdx.x) >> 5;
    const int lane = threadIdx.x & 31;
    if (wave >= EE) return;

    const int s = src[wave];
    const int d = dst[wave];
    const int t = etype[wave];

    const float4 v = ((const float4*)(x + (size_t)s * FF))[lane];

    float* ar = agg + ((size_t)d * RR + t) * FF + lane * 4;
    atomicAdd(ar + 0, v.x);
    atomicAdd(ar + 1, v.y);
    atomicAdd(ar + 2, v.z);
    atomicAdd(ar + 3, v.w);
    if (lane == 0) atomicAdd(cnt + (size_t)d * RR + t, 1.0f);
}

// ---------------------------------------------------------------------------
// Phase 2: fused GEMM
//   out = relu( sum_r (agg_r / max(cnt_r,1)) @ W_r  +  x @ root  +  bias )
// i.e. a [N, 5*128] x [5*128, 128] GEMM. K is processed in 10 chunks of 64
// through LDS. fp32 precision preserved via V_WMMA_F32_16X16X4_F32.
// Block = 256 threads = 8 waves. M-tile = 64 rows. Wave w owns output columns
// [16w, 16w+16) and iterates 4 M-subtiles of 16 rows.
// ---------------------------------------------------------------------------
#define MT   64
#define KC   64
#define ASTR (KC + 4)   // LDS padding to avoid bank conflicts
#define BSTR (FF + 4)

__global__ __launch_bounds__(256)
void rgcn_gemm(const float* __restrict__ xin,   // [N, F]
               const float* __restrict__ agg,   // [N, R, F]
               const float* __restrict__ cnt,   // [N, R]
               const float* __restrict__ Wl,    // [R, F, F]  (this layer)
               const float* __restrict__ rootl, // [F, F]
               const float* __restrict__ biasl, // [F]
               float* __restrict__ out,         // [N, F]
               int nrows)
{
    __shared__ float As[MT * ASTR];
    __shared__ float Bs[KC * BSTR];

    const int tid   = threadIdx.x;
    const int wave  = tid >> 5;
    const int lane  = tid & 31;
    const int l16   = lane & 15;
    const int lhalf = lane >> 4;        // selects K MSB for A/B fragments
    const int m0    = blockIdx.x * MT;
    const int o0    = wave * 16;

    v8f acc[4];
    const v8f vzero = {0.f, 0.f, 0.f, 0.f, 0.f, 0.f, 0.f, 0.f};
#pragma unroll
    for (int i = 0; i < 4; ++i) acc[i] = vzero;

    // Staging role: 4 threads per row, thread covers a contiguous span.
    const int rowA = tid >> 2;          // 0..63  (A tile row / B chunk row)
    const int q    = tid & 3;

    for (int c = 0; c < 10; ++c) {
        // ---- stage A chunk (64 rows x 64 K) scaled by 1/max(cnt,1) ----
        {
            const int  n     = m0 + rowA;
            const bool valid = (n < nrows);
            const float* srcA = xin;    // dummy init
            float scale = 1.0f;
            if (c < 8) {
                const int r  = c >> 1;
                const int f0 = (c & 1) * KC;
                if (valid) {
                    const float cc = cnt[(size_t)n * RR + r];
                    scale = 1.0f / fmaxf(cc, 1.0f);
                    srcA  = agg + ((size_t)n * RR + r) * FF + f0;
                }
            } else {
                const int f0 = (c - 8) * KC;
                if (valid) srcA = xin + (size_t)n * FF + f0;
            }
            float* dstA = As + rowA * ASTR + q * 16;
#pragma unroll
            for (int j = 0; j < 4; ++j) {
                float4 v = make_float4(0.f, 0.f, 0.f, 0.f);
                if (valid) v = *(const float4*)(srcA + q * 16 + j * 4);
                dstA[j * 4 + 0] = v.x * scale;
                dstA[j * 4 + 1] = v.y * scale;
                dstA[j * 4 + 2] = v.z * scale;
                dstA[j * 4 + 3] = v.w * scale;
            }
        }
        // ---- stage B chunk (64 K x 128 O) ----
        {
            const float* srcB;
            if (c < 8) {
                const int r  = c >> 1;
                const int f0 = (c & 1) * KC;
                srcB = Wl + ((size_t)r * FF + (f0 + rowA)) * FF;
            } else {
                const int f0 = (c - 8) * KC;
                srcB = rootl + (size_t)(f0 + rowA) * FF;
            }
            float* dstB = Bs + rowA * BSTR + q * 32;
#pragma unroll
            for (int j = 0; j < 8; ++j) {
                const float4 v = *(const float4*)(srcB + q * 32 + j * 4);
                dstB[j * 4 + 0] = v.x;
                dstB[j * 4 + 1] = v.y;
                dstB[j * 4 + 2] = v.z;
                dstB[j * 4 + 3] = v.w;
            }
        }
        __syncthreads();

        // ---- WMMA over this K chunk: 16 steps of K=4, 4 M-subtiles ----
#pragma unroll 4
        for (int kb = 0; kb < KC; kb += 4) {
            const int kk = kb + 2 * lhalf;   // lane-half = K MSB (+2)
            v2f b;
            b.x = Bs[(kk + 0) * BSTR + o0 + l16];
            b.y = Bs[(kk + 1) * BSTR + o0 + l16];
#pragma unroll
            for (int mt = 0; mt < 4; ++mt) {
                const int m = mt * 16 + l16;
                v2f a;
                a.x = As[m * ASTR + kk + 0];
                a.y = As[m * ASTR + kk + 1];
                acc[mt] = __builtin_amdgcn_wmma_f32_16x16x4_f32(
                    /*neg_a=*/false, a, /*neg_b=*/false, b,
                    /*c_mod=*/(short)0, acc[mt],
                    /*reuse_a=*/false, /*reuse_b=*/false);
            }
        }
        __syncthreads();
    }

    // ---- epilogue: + bias, ReLU, store (C/D layout: row = v + 8*lanehalf) ----
    const int   col = o0 + l16;
    const float bv  = biasl[col];
#pragma unroll
    for (int mt = 0; mt < 4; ++mt) {
#pragma unroll
        for (int v = 0; v < 8; ++v) {
            const int row = m0 + mt * 16 + v + 8 * lhalf;
            if (row < nrows) {
                float val = acc[mt][v] + bv;
                out[(size_t)row * FF + col] = val > 0.f ? val : 0.f;
            }
        }
    }
}

// ---------------------------------------------------------------------------
extern "C" void kernel_launch(void* const* d_in, const int* in_sizes, int n_in,
                              void* d_out, int out_size, void* d_ws, size_t ws_size,
                              hipStream_t stream)
{
    const float* x       = (const float*)d_in[0];
    const int*   eidx    = (const int*)d_in[1];      // [2, E]
    const int*   etype   = (const int*)d_in[2];      // [E]
    const float* weights = (const float*)d_in[3];    // [L, R, F, F]
    const float* roots   = (const float*)d_in[4];    // [L, F, F]
    const float* biases  = (const float*)d_in[5];    // [L, F]

    const int* src = eidx;
    const int* dst = eidx + EE;

    // workspace layout: agg [N*R*F] | cnt [N*R] | h1 [N*F]
    float* agg = (float*)d_ws;
    float* cnt = agg + (size_t)NN * RR * FF;
    float* h1  = cnt + (size_t)NN * RR;

    const int scatter_blocks = (EE * 32) / 256;            // 1 wave per edge
    const int gemm_blocks    = (NN + MT - 1) / MT;

    const float* layer_in = x;
    for (int l = 0; l < LL; ++l) {
        hipMemsetAsync(agg, 0, (size_t)NN * RR * FF * sizeof(float), stream);
        hipMemsetAsync(cnt, 0, (size_t)NN * RR * sizeof(float), stream);

        rgcn_scatter<<<scatter_blocks, 256, 0, stream>>>(
            layer_in, src, dst, etype, agg, cnt);

        const float* Wl    = weights + (size_t)l * RR * FF * FF;
        const float* rootl = roots   + (size_t)l * FF * FF;
        const float* biasl = biases  + (size_t)l * FF;
        float* layer_out = (l == LL - 1) ? (float*)d_out : h1;

        rgcn_gemm<<<gemm_blocks, 256, 0, stream>>>(
            layer_in, agg, cnt, Wl, rootl, biasl, layer_out, NN);

        layer_in = layer_out;
    }
}